// DensePoint_47906065219820
// MI455X (gfx1250) — compile-verified
//
#include <hip/hip_runtime.h>
#include <cstddef>
#include <cstdint>

// ===================== CDNA5 WMMA helpers (gfx1250, wave32) =====================
typedef float v2f __attribute__((ext_vector_type(2)));
typedef float v8f __attribute__((ext_vector_type(8)));

// V_WMMA_F32_16X16X4_F32: D(16x16 f32) = A(16x4 f32) * B(4x16 f32) + C
// A layout (2 VGPRs): lanes 0-15 hold row M=lane, K={0,1} in {x,y}; lanes 16-31: K={2,3}.
// B layout assumed symmetric: lanes 0-15 col N=lane, K={0,1}; lanes 16-31: K={2,3}.
// C/D layout: VGPR v: lanes 0-15 -> (M=v, N=lane); lanes 16-31 -> (M=v+8, N=lane-16).
__device__ __forceinline__ v8f wmma4(v2f a, v2f b, v8f c) {
  return __builtin_amdgcn_wmma_f32_16x16x4_f32(false, a, false, b, (short)0, c,
                                               false, false);
}

// unguarded 8-byte vector load (callers guarantee 8B alignment: even k, even row strides)
__device__ __forceinline__ v2f ld2(const float* p) { return *(const v2f*)p; }

__device__ __forceinline__ float max8(v8f a) {
  float m = a[0];
#pragma unroll
  for (int i = 1; i < 8; ++i) m = fmaxf(m, a[i]);
  return m;
}

// ===================== Farthest point sampling =====================
__global__ void fps_kernel(const float* __restrict__ pts, int n, int npoint,
                           int* __restrict__ out) {
  __shared__ float dists[2048];
  __shared__ float rv[256];
  __shared__ int ri[256];
  __shared__ int lastSh;
  const int b = blockIdx.x, tid = threadIdx.x, bd = blockDim.x;
  const float* X = pts + (size_t)b * n * 3;
  int* O = out + (size_t)b * npoint;
  for (int i = tid; i < n; i += bd) dists[i] = 1e10f;
  if (tid == 0) { O[0] = 0; lastSh = 0; }
  __syncthreads();
  int last = 0;
  for (int it = 1; it < npoint; ++it) {
    float lx = X[last * 3], ly = X[last * 3 + 1], lz = X[last * 3 + 2];
    float bv = -1.f;
    int bi = 0;
    for (int i = tid; i < n; i += bd) {
      float dx = X[i * 3] - lx, dy = X[i * 3 + 1] - ly, dz = X[i * 3 + 2] - lz;
      float d = fminf(dists[i], dx * dx + dy * dy + dz * dz);
      dists[i] = d;
      if (d > bv) { bv = d; bi = i; }
    }
    rv[tid] = bv; ri[tid] = bi;
    __syncthreads();
    for (int off = bd >> 1; off > 0; off >>= 1) {
      if (tid < off) {
        float v2 = rv[tid + off]; int i2 = ri[tid + off];
        if (v2 > rv[tid] || (v2 == rv[tid] && i2 < ri[tid])) { rv[tid] = v2; ri[tid] = i2; }
      }
      __syncthreads();
    }
    if (tid == 0) { O[it] = ri[0]; lastSh = ri[0]; }
    __syncthreads();
    last = lastSh;
  }
}

// ===================== Ball query (+ center gather) =====================
__global__ void ballq_kernel(const float* __restrict__ pts,
                             const int* __restrict__ fidx,
                             float* __restrict__ ctr, int* __restrict__ idx,
                             int n, int S, int nsample, float r2, int BS) {
  int t = blockIdx.x * blockDim.x + threadIdx.x;
  if (t >= BS) return;
  int b = t / S, si = t - b * S;
  int c = fidx[(size_t)b * S + si];
  const float* X = pts + (size_t)b * n * 3;
  float cx = X[c * 3], cy = X[c * 3 + 1], cz = X[c * 3 + 2];
  ctr[(size_t)t * 3 + 0] = cx;
  ctr[(size_t)t * 3 + 1] = cy;
  ctr[(size_t)t * 3 + 2] = cz;
  int* o = idx + (size_t)t * nsample;
  int cnt = 0, first = -1;
  for (int j = 0; j < n; ++j) {
    float dx = X[j * 3] - cx, dy = X[j * 3 + 1] - cy, dz = X[j * 3 + 2] - cz;
    if (dx * dx + dy * dy + dz * dz < r2) {
      if (first < 0) first = j;
      if (cnt < nsample) o[cnt] = j;
      ++cnt;
    }
  }
  if (first < 0) first = 0;
  for (int k = cnt; k < nsample; ++k) o[k] = first;
  o[0] = c;
}

// ===================== ppool: gather -> GEMM -> bn_relu -> max over K =====================
// one workgroup (128 thr / 4 waves) per center.
// LDS: A (K x cinp, zero-padded) and W^T (coutp x cinp, zero-padded) staged once;
// inner loop is two unguarded ds_load_b64 + v_wmma (no EXEC churn).
__global__ void ppool_kernel(const float* __restrict__ pts, const float* __restrict__ fin,
                             const int* __restrict__ idx, const float* __restrict__ ctr,
                             const float* __restrict__ W, const float* __restrict__ bias,
                             const float* __restrict__ g, const float* __restrict__ be,
                             float* __restrict__ fout, int n, int S, int K, int cin,
                             int cinp, int cout, int ntiles) {
  extern __shared__ float sh[];
  float* Ash = sh;                       // K * cinp
  float* Bsh = sh + K * cinp;            // (ntiles*16) * cinp, transposed [col][k]
  const int bs = blockIdx.x, b = bs / S, tid = threadIdx.x;
  const int cf = cin - 3;
  const int coutp = ntiles * 16;
  const int* id = idx + (size_t)bs * K;
  const float* c3 = ctr + (size_t)bs * 3;
  for (int e = tid; e < K * cinp; e += blockDim.x) {
    int r = e / cinp, ch = e - r * cinp;
    float v = 0.f;
    if (ch < cin) {
      int j = id[r];
      v = (ch < 3) ? (pts[((size_t)b * n + j) * 3 + ch] - c3[ch])
                   : fin[((size_t)b * n + j) * cf + (ch - 3)];
    }
    Ash[r * cinp + ch] = v;
  }
  for (int e = tid; e < coutp * cinp; e += blockDim.x) {
    int c = e / cinp, k = e - c * cinp;
    Bsh[e] = (k < cin && c < cout) ? W[(size_t)k * cout + c] : 0.f;
  }
  __syncthreads();
  const int wave = tid >> 5, lane = tid & 31, nw = blockDim.x >> 5;
  const int l15 = lane & 15;
  const int khalf = (lane >> 4) << 1;
  for (int nt = wave; nt < ntiles; nt += nw) {
    int col = nt * 16 + l15;
    const float* Bcol = Bsh + (size_t)col * cinp;
    float colmax = -3.4e38f;
    for (int mt = 0; mt < K; mt += 16) {
      const float* Arow = Ash + (size_t)(mt + l15) * cinp;
      v8f acc = {0.f, 0.f, 0.f, 0.f, 0.f, 0.f, 0.f, 0.f};
      for (int k0 = 0; k0 < cinp; k0 += 4) {
        v2f a = ld2(Arow + k0 + khalf);
        v2f bb = ld2(Bcol + k0 + khalf);
        acc = wmma4(a, bb, acc);
      }
      colmax = fmaxf(colmax, max8(acc));
    }
    colmax = fmaxf(colmax, __shfl_xor(colmax, 16, 32));  // combine M halves
    if (lane < 16 && col < cout) {
      float o = colmax + bias[col];
      fout[(size_t)bs * cout + col] = fmaxf(o * g[col] + be[col], 0.f);
    }
  }
}

// ===================== pconv: gather(+bn_cin) -> grouped GEMM -> max -> psi -> concat(+bn_cat) =====================
__global__ void pconv_kernel(const float* __restrict__ pts, const float* __restrict__ fin,
                             const int* __restrict__ idx, const float* __restrict__ ctr,
                             const float* __restrict__ Wphi, const float* __restrict__ bphi,
                             const float* __restrict__ gphi, const float* __restrict__ bephi,
                             const float* __restrict__ Wpsi, const float* __restrict__ bpsi,
                             const float* __restrict__ gcin, const float* __restrict__ becin,
                             const float* __restrict__ gcat, const float* __restrict__ becat,
                             float* __restrict__ fout, int n, int S, int cin, int cpg) {
  __shared__ float Ash[16 * 144];
  __shared__ float Wsh[96 * 72];   // W_phi transposed: [col 0..95][k 0..cpg-1]
  __shared__ float hbuf[96];
  const int bs = blockIdx.x, b = bs / S, tid = threadIdx.x;
  const int cf = cin - 3;
  const int* id = idx + (size_t)bs * 16;
  const float* c3 = ctr + (size_t)bs * 3;
  for (int e = tid; e < 16 * cin; e += blockDim.x) {
    int r = e / cin, ch = e - r * cin;
    int j = id[r];
    float v = (ch < 3) ? (pts[((size_t)b * n + j) * 3 + ch] - c3[ch])
                       : fin[((size_t)b * n + j) * cf + (ch - 3)];
    if (gcin) v = fmaxf(v * gcin[ch] + becin[ch], 0.f);
    Ash[r * cin + ch] = v;
  }
  for (int e = tid; e < 96 * cpg; e += blockDim.x) {
    int c = e / cpg, k = e - c * cpg;
    int gi = c / 48;
    Wsh[e] = Wphi[((size_t)gi * cpg + k) * 48 + (c - gi * 48)];
  }
  __syncthreads();
  const int wave = tid >> 5, lane = tid & 31, nw = blockDim.x >> 5;
  const int l15 = lane & 15;
  const int khalf = (lane >> 4) << 1;
  for (int nt = wave; nt < 6; nt += nw) {           // 6 N-tiles of 16 cols = 96
    int gidx = nt / 3;                              // group id, uniform per wave
    int col = nt * 16 + l15;
    const float* Arow = Ash + (size_t)l15 * cin + gidx * cpg;
    const float* Bcol = Wsh + (size_t)col * cpg;
    v8f acc = {0.f, 0.f, 0.f, 0.f, 0.f, 0.f, 0.f, 0.f};
    for (int k0 = 0; k0 < cpg; k0 += 4) {           // cpg in {48,60,72}, /4 exact
      v2f a = ld2(Arow + k0 + khalf);
      v2f bb = ld2(Bcol + k0 + khalf);
      acc = wmma4(a, bb, acc);
    }
    float m = max8(acc);
    m = fmaxf(m, __shfl_xor(m, 16, 32));
    if (lane < 16) {
      float o = m + bphi[col];
      hbuf[col] = fmaxf(o * gphi[col] + bephi[col], 0.f);
    }
  }
  __syncthreads();
  const int cout_tot = cf + 24;
  for (int o = tid; o < cout_tot; o += blockDim.x) {
    float v;
    if (o < cf) {
      v = Ash[3 + o];                               // row 0 (center), channels 3..
    } else {
      int oo = o - cf;
      float s = bpsi[oo];
      for (int c = 0; c < 96; ++c) s += hbuf[c] * Wpsi[c * 24 + oo];
      v = s;
    }
    if (gcat) v = fmaxf(v * gcat[o] + becat[o], 0.f);
    fout[(size_t)bs * cout_tot + o] = v;
  }
}

// ===================== concat [xyz | feats] -> (B*128, 168) =====================
__global__ void cat168_kernel(const float* __restrict__ xyz, const float* __restrict__ f,
                              float* __restrict__ out, int rows) {
  int t = blockIdx.x * blockDim.x + threadIdx.x;
  if (t >= rows * 168) return;
  int r = t / 168, c = t - r * 168;
  out[t] = (c < 3) ? xyz[(size_t)r * 3 + c] : f[(size_t)r * 165 + (c - 3)];
}

// ===================== m6: GEMM(128x168 @ 168x512) + bn_relu + max over 128 rows =====================
// one wave per (batch, 16-col tile); W tile staged transposed in LDS, A via global b64.
__global__ void m6_kernel(const float* __restrict__ A, const float* __restrict__ W,
                          const float* __restrict__ bias, const float* __restrict__ g,
                          const float* __restrict__ be, float* __restrict__ out) {
  __shared__ float Bsh[16 * 168];  // [col 0..15][k 0..167]
  const int nt = blockIdx.x & 31, b = blockIdx.x >> 5;
  const int lane = threadIdx.x & 31;
  const int l15 = lane & 15;
  const int khalf = (lane >> 4) << 1;
  for (int e = threadIdx.x; e < 16 * 168; e += 32) {
    int c = e / 168, k = e - c * 168;
    Bsh[e] = W[(size_t)k * 512 + nt * 16 + c];
  }
  __syncthreads();
  const float* Ab = A + (size_t)b * 128 * 168;
  const float* Bcol = Bsh + (size_t)l15 * 168;
  float colmax = -3.4e38f;
  for (int mt = 0; mt < 128; mt += 16) {
    const float* Arow = Ab + (size_t)(mt + l15) * 168;
    v8f acc = {0.f, 0.f, 0.f, 0.f, 0.f, 0.f, 0.f, 0.f};
    for (int k0 = 0; k0 < 168; k0 += 4) {
      v2f a = ld2(Arow + k0 + khalf);
      v2f bb = ld2(Bcol + k0 + khalf);
      acc = wmma4(a, bb, acc);
    }
    colmax = fmaxf(colmax, max8(acc));
  }
  colmax = fmaxf(colmax, __shfl_xor(colmax, 16, 32));
  if (lane < 16) {
    int col = nt * 16 + l15;
    float o = colmax + bias[col];
    out[(size_t)b * 512 + col] = fmaxf(o * g[col] + be[col], 0.f);
  }
}

// ===================== generic fc GEMM (+ optional bn_relu) =====================
// one wave per 16x16 output tile; W tile staged transposed+padded in dynamic LDS.
// Requires K % 4 == 0, M % 16 == 0 (true for all fc layers here).
__global__ void gemm_kernel(const float* __restrict__ A, const float* __restrict__ W,
                            const float* __restrict__ bias, const float* __restrict__ g,
                            const float* __restrict__ be, float* __restrict__ out,
                            int M, int K, int N, int mtiles) {
  extern __shared__ float Bsh[];  // [c 0..15][k 0..K-1], zero-padded cols
  const int mt = blockIdx.x % mtiles, nt = blockIdx.x / mtiles;
  const int lane = threadIdx.x & 31;
  const int l15 = lane & 15;
  const int khalf = (lane >> 4) << 1;
  for (int e = threadIdx.x; e < 16 * K; e += 32) {
    int c = e / K, k = e - c * K;
    int col = nt * 16 + c;
    Bsh[e] = (col < N) ? W[(size_t)k * N + col] : 0.f;
  }
  __syncthreads();
  const float* Arow = A + (size_t)(mt * 16 + l15) * K;
  const float* Bcol = Bsh + (size_t)l15 * K;
  v8f acc = {0.f, 0.f, 0.f, 0.f, 0.f, 0.f, 0.f, 0.f};
  for (int k0 = 0; k0 < K; k0 += 4) {
    v2f a = ld2(Arow + k0 + khalf);
    v2f bb = ld2(Bcol + k0 + khalf);
    acc = wmma4(a, bb, acc);
  }
  int col = nt * 16 + l15;
  if (col < N) {
    float bc = bias[col];
#pragma unroll
    for (int v = 0; v < 8; ++v) {
      int row = mt * 16 + v + ((lane >> 4) << 3);
      float o = acc[v] + bc;
      if (g) o = fmaxf(o * g[col] + be[col], 0.f);
      out[(size_t)row * N + col] = o;
    }
  }
}

// ===================== host orchestration =====================
extern "C" void kernel_launch(void* const* d_in, const int* in_sizes, int n_in,
                              void* d_out, int out_size, void* d_ws, size_t ws_size,
                              hipStream_t stream) {
  (void)in_sizes; (void)n_in; (void)out_size; (void)ws_size;
  const int B = 32, N = 2048;
  const float* pc = (const float*)d_in[0];
  auto F = [&](int i) { return (const float*)d_in[i]; };
  // m1: 1-4   m2: 5-8
  // m3: 9 Wphi 10 bphi 11 gphi 12 bephi 13 Wpsi 14 bpsi
  // m4: 15-20 + 21 gcin 22 becin
  // m5: 23-28 + 29 gcin 30 becin 31 gcat 32 becat
  // m6: 33-36  fc1: 37-40  fc2: 41-44  fc3: 45-46
  char* w = (char*)d_ws;
  size_t off = 0;
  auto AL = [&](size_t bytes) -> void* {
    void* p = w + off;
    off = (off + bytes + 255) & ~(size_t)255;
    return p;
  };
  int* fidx1 = (int*)AL((size_t)B * 512 * 4);
  float* nx1 = (float*)AL((size_t)B * 512 * 3 * 4);
  int* idx1 = (int*)AL((size_t)B * 512 * 64 * 4);
  float* f1 = (float*)AL((size_t)B * 512 * 96 * 4);
  int* fidx2 = (int*)AL((size_t)B * 128 * 4);
  float* nx2 = (float*)AL((size_t)B * 128 * 3 * 4);
  int* idx2 = (int*)AL((size_t)B * 128 * 64 * 4);
  float* f2 = (float*)AL((size_t)B * 128 * 93 * 4);
  int* fidx3 = (int*)AL((size_t)B * 128 * 4);
  float* nx3 = (float*)AL((size_t)B * 128 * 3 * 4);
  int* idx3 = (int*)AL((size_t)B * 128 * 16 * 4);
  float* f3 = (float*)AL((size_t)B * 128 * 117 * 4);
  int* fidx4 = (int*)AL((size_t)B * 128 * 4);
  float* nx4 = (float*)AL((size_t)B * 128 * 3 * 4);
  int* idx4 = (int*)AL((size_t)B * 128 * 16 * 4);
  float* f4 = (float*)AL((size_t)B * 128 * 141 * 4);
  int* fidx5 = (int*)AL((size_t)B * 128 * 4);
  float* nx5 = (float*)AL((size_t)B * 128 * 3 * 4);
  int* idx5 = (int*)AL((size_t)B * 128 * 16 * 4);
  float* f5 = (float*)AL((size_t)B * 128 * 165 * 4);
  float* cat6 = (float*)AL((size_t)B * 128 * 168 * 4);
  float* feat6 = (float*)AL((size_t)B * 512 * 4);
  float* h1 = (float*)AL((size_t)B * 512 * 4);
  float* h2 = (float*)AL((size_t)B * 256 * 4);

  const int BS1 = B * 512, BS2 = B * 128;
  // ---- m1 ----
  fps_kernel<<<B, 256, 0, stream>>>(pc, N, 512, fidx1);
  ballq_kernel<<<(BS1 + 127) / 128, 128, 0, stream>>>(pc, fidx1, nx1, idx1, N, 512, 64,
                                                      0.0625f, BS1);
  ppool_kernel<<<BS1, 128, (64 * 4 + 96 * 4) * sizeof(float), stream>>>(
      pc, nullptr, idx1, nx1, F(1), F(2), F(3), F(4), f1, N, 512, 64, 3, 4, 96, 6);
  // ---- m2 ----
  fps_kernel<<<B, 256, 0, stream>>>(nx1, 512, 128, fidx2);
  ballq_kernel<<<(BS2 + 127) / 128, 128, 0, stream>>>(nx1, fidx2, nx2, idx2, 512, 128, 64,
                                                      0.1024f, BS2);
  ppool_kernel<<<BS2, 128, (64 * 100 + 96 * 100) * sizeof(float), stream>>>(
      nx1, f1, idx2, nx2, F(5), F(6), F(7), F(8), f2, 512, 128, 64, 99, 100, 93, 6);
  // ---- m3 (after_pool) ----
  fps_kernel<<<B, 256, 0, stream>>>(nx2, 128, 128, fidx3);
  ballq_kernel<<<(BS2 + 127) / 128, 128, 0, stream>>>(nx2, fidx3, nx3, idx3, 128, 128, 16,
                                                      0.1521f, BS2);
  pconv_kernel<<<BS2, 128, 0, stream>>>(nx2, f2, idx3, nx3, F(9), F(10), F(11), F(12),
                                        F(13), F(14), nullptr, nullptr, nullptr, nullptr,
                                        f3, 128, 128, 96, 48);
  // ---- m4 (bn_cin) ----
  fps_kernel<<<B, 256, 0, stream>>>(nx3, 128, 128, fidx4);
  ballq_kernel<<<(BS2 + 127) / 128, 128, 0, stream>>>(nx3, fidx4, nx4, idx4, 128, 128, 16,
                                                      0.1521f, BS2);
  pconv_kernel<<<BS2, 128, 0, stream>>>(nx3, f3, idx4, nx4, F(15), F(16), F(17), F(18),
                                        F(19), F(20), F(21), F(22), nullptr, nullptr,
                                        f4, 128, 128, 120, 60);
  // ---- m5 (bn_cin + bn_cat) ----
  fps_kernel<<<B, 256, 0, stream>>>(nx4, 128, 128, fidx5);
  ballq_kernel<<<(BS2 + 127) / 128, 128, 0, stream>>>(nx4, fidx5, nx5, idx5, 128, 128, 16,
                                                      0.1521f, BS2);
  pconv_kernel<<<BS2, 128, 0, stream>>>(nx4, f4, idx5, nx5, F(23), F(24), F(25), F(26),
                                        F(27), F(28), F(29), F(30), F(31), F(32),
                                        f5, 128, 128, 144, 72);
  // ---- m6: concat -> GEMM+bn_relu+max over points ----
  cat168_kernel<<<(B * 128 * 168 + 255) / 256, 256, 0, stream>>>(nx5, f5, cat6, B * 128);
  m6_kernel<<<B * 32, 32, 0, stream>>>(cat6, F(33), F(34), F(35), F(36), feat6);
  // ---- fc head ----
  gemm_kernel<<<2 * 32, 32, 16 * 512 * sizeof(float), stream>>>(
      feat6, F(37), F(38), F(39), F(40), h1, 32, 512, 512, 2);
  gemm_kernel<<<2 * 16, 32, 16 * 512 * sizeof(float), stream>>>(
      h1, F(41), F(42), F(43), F(44), h2, 32, 512, 256, 2);
  gemm_kernel<<<2 * 1, 32, 16 * 256 * sizeof(float), stream>>>(
      h2, F(45), F(46), nullptr, nullptr, (float*)d_out, 32, 256, 7, 2);
}